// _DCN_Module_39264591020165
// MI455X (gfx1250) — compile-verified
//
#include <hip/hip_runtime.h>

typedef float v2f __attribute__((ext_vector_type(2)));
typedef float v8f __attribute__((ext_vector_type(8)));

#define N_SAMPLES 500000
#define DIMS      64
#define KCLUST    256
// ws layout (float units): [0,16384) seg_sum; [16384,16640) seg_cnt (int32);
//                          [16640,16896) c2;  [16896] loss accumulator
#define WS_SEGSUM 0
#define WS_SEGCNT 16384
#define WS_C2     16640
#define WS_LOSS   16896

__global__ void dcn_prep(const float* __restrict__ centers, float* __restrict__ ws) {
    int i = blockIdx.x * blockDim.x + threadIdx.x;
    if (i < KCLUST * DIMS) ws[WS_SEGSUM + i] = 0.f;
    if (i < KCLUST) {
        ((int*)ws)[WS_SEGCNT + i] = 0;
        const float* c = centers + i * DIMS;
        float s = 0.f;
#pragma unroll 8
        for (int d = 0; d < DIMS; ++d) s += c[d] * c[d];
        ws[WS_C2 + i] = s;
    }
    if (i == 0) ws[WS_LOSS] = 0.f;
}

__global__ void __launch_bounds__(256)
dcn_main(const float* __restrict__ embedded,
         const float* __restrict__ centers,
         float* __restrict__ ws,
         int* __restrict__ labels) {
    __shared__ float s_centers[KCLUST * DIMS];   // 64 KB, all centers resident
    {
        const float4* src = reinterpret_cast<const float4*>(centers);
        float4*       dst = reinterpret_cast<float4*>(s_centers);
        for (int i = threadIdx.x; i < KCLUST * DIMS / 4; i += blockDim.x) dst[i] = src[i];
    }
    __syncthreads();

    const float* c2p      = ws + WS_C2;
    float*       seg_sum  = ws + WS_SEGSUM;
    int*         seg_cnt  = (int*)ws + WS_SEGCNT;
    float*       loss_acc = ws + WS_LOSS;

    const int lane  = threadIdx.x & 31;
    const int wave  = threadIdx.x >> 5;
    const int wpb   = blockDim.x >> 5;
    const int gwave = blockIdx.x * wpb + wave;
    const int nwav  = gridDim.x * wpb;
    const int ntile = N_SAMPLES / 16;            // 31250 exactly

    const int lrow  = lane & 15;                 // sample-in-tile (B) / cluster-in-tile (A)
    const int hi    = lane >> 4;                 // half-wave select
    const int dbase = hi * 2;                    // K-offset within each group of 4 dims

    for (int tile = gwave; tile < ntile; tile += nwav) {
        const int n0 = tile * 16;

        // B operand: 16 samples x 64 dims; per lane 16 x float2 (K={0,1} lo / {2,3} hi)
        v2f   xv[16];
        float x2 = 0.f;
        const float* xrow = embedded + (size_t)(n0 + lrow) * DIMS + dbase;
#pragma unroll
        for (int d = 0; d < 16; ++d) {
            xv[d] = *reinterpret_cast<const v2f*>(xrow + 4 * d);
            x2 += xv[d][0] * xv[d][0] + xv[d][1] * xv[d][1];
        }
        x2 += __shfl_xor(x2, 16, 32);            // full ||x||^2 per sample

        float bestv = 3.402823466e38f;
        int   besti = 0;

#pragma unroll 1
        for (int t = 0; t < 16; ++t) {           // 16 cluster tiles of 16
            v8f acc = {0.f, 0.f, 0.f, 0.f, 0.f, 0.f, 0.f, 0.f};
            const float* crow = s_centers + (t * 16 + lrow) * DIMS + dbase;
#pragma unroll
            for (int d = 0; d < 16; ++d) {       // K = 64 in steps of 4
                v2f a = *reinterpret_cast<const v2f*>(crow + 4 * d);
                acc = __builtin_amdgcn_wmma_f32_16x16x4_f32(
                          false, a, false, xv[d], (short)0, acc, false, false);
            }
            // lane holds scores for clusters cbase..cbase+7, its own sample
            const int cbase = t * 16 + hi * 8;
            float4 c2a = *reinterpret_cast<const float4*>(c2p + cbase);
            float4 c2b = *reinterpret_cast<const float4*>(c2p + cbase + 4);
            float  cv[8] = {c2a.x, c2a.y, c2a.z, c2a.w, c2b.x, c2b.y, c2b.z, c2b.w};
#pragma unroll
            for (int r = 0; r < 8; ++r) {
                float v = cv[r] - 2.f * acc[r];  // dist^2 - ||x||^2 (row-constant)
                if (v < bestv) { bestv = v; besti = cbase + r; }
            }
        }
        // merge the two half-wave cluster ranges (tie -> lower index)
        {
            float ov = __shfl_xor(bestv, 16, 32);
            int   oi = __shfl_xor(besti, 16, 32);
            if (ov < bestv || (ov == bestv && oi < besti)) { bestv = ov; besti = oi; }
        }
        if (lane < 16) labels[n0 + lrow] = besti;

        // loss: min dist^2 = ||x||^2 + bestv ; wave-reduce then one atomic
        float li = (lane < 16) ? (x2 + bestv) : 0.f;
#pragma unroll
        for (int off = 16; off > 0; off >>= 1) li += __shfl_xor(li, off, 32);
        if (lane == 0) unsafeAtomicAdd(loss_acc, li);

        // segment sums: each lane owns dims {dbase, dbase+1} mod 4 of its sample
        float* sb = seg_sum + (size_t)besti * DIMS + dbase;
#pragma unroll
        for (int d = 0; d < 16; ++d) {
            unsafeAtomicAdd(sb + 4 * d,     xv[d][0]);
            unsafeAtomicAdd(sb + 4 * d + 1, xv[d][1]);
        }
        if (lane < 16) atomicAdd(seg_cnt + besti, 1);
    }
}

__global__ void dcn_finalize(const float* __restrict__ centers,
                             const int* __restrict__ counts,
                             const float* __restrict__ ws,
                             float* __restrict__ out) {
    int i = blockIdx.x * blockDim.x + threadIdx.x;
    const int* seg_cnt = (const int*)ws + WS_SEGCNT;
    if (i < KCLUST * DIMS) {
        int   c  = i >> 6;
        float oc = (float)counts[c];
        float nw = (float)(counts[c] + seg_cnt[c]);
        out[N_SAMPLES + 1 + i] = (oc * centers[i] + ws[WS_SEGSUM + i]) / nw;
    }
    if (i < KCLUST)
        ((int*)out)[N_SAMPLES + 1 + KCLUST * DIMS + i] = counts[i] + seg_cnt[i];
    if (i == 0)
        out[N_SAMPLES] = ws[WS_LOSS] / (float)N_SAMPLES;
}

extern "C" void kernel_launch(void* const* d_in, const int* in_sizes, int n_in,
                              void* d_out, int out_size, void* d_ws, size_t ws_size,
                              hipStream_t stream) {
    const float* embedded = (const float*)d_in[0];   // [500000,64] f32
    const float* centers  = (const float*)d_in[1];   // [256,64]    f32
    const int*   counts   = (const int*)d_in[2];     // [256]       i32
    float* out = (float*)d_out;                      // labels|loss|new_centers|new_counts
    float* ws  = (float*)d_ws;

    dcn_prep<<<64, 256, 0, stream>>>(centers, ws);
    dcn_main<<<512, 256, 0, stream>>>(embedded, centers, ws, (int*)d_out);
    dcn_finalize<<<64, 256, 0, stream>>>(centers, counts, ws, out);
}